// DecomposedEmbedding_79628693667922
// MI455X (gfx1250) — compile-verified
//
#include <hip/hip_runtime.h>
#include <stdint.h>

// DecomposedEmbedding fused gather for MI455X (gfx1250).
//
// out[t, d] = sw[idx,d]*sigmoid(mask[idx]) + aw[idx,d]
//           + atten[0]*from_kb[idx,d,0] + atten[1]*from_kb[idx,d,1]
//
// Pure bandwidth-bound row gather (~42 MB total traffic). One wave32 per
// token stage; rows are staged LDS-side with the CDNA5 async copy engine
// (global_load_async_to_lds_b128 / ASYNCcnt) and double-buffered so the
// next token's 2KB of rows is in flight while the current one is consumed.

#define WAVES_PER_BLOCK 8
#define STAGE_BYTES     2048   // 512 (sw row) + 512 (aw row) + 1024 (from_kb row)
#define EMBED_DIM       128
#define ROW_BYTES_SWAW  512u   // 128 * 4B
#define ROW_BYTES_FKB   1024u  // 128 * 2 * 4B

__device__ __forceinline__ void async_issue_row(uint32_t lds_sw, uint32_t lds_aw,
                                                uint32_t lds_fk,
                                                uint64_t g_sw, uint64_t g_aw,
                                                uint64_t g_fk) {
  // Per-lane 16B chunks: sw row + aw row (one B128 each), from_kb row
  // (two B128; offset:16 applies to both LDS and global address per ISA).
  asm volatile("global_load_async_to_lds_b128 %0, %1, off"
               :: "v"(lds_sw), "v"(g_sw) : "memory");
  asm volatile("global_load_async_to_lds_b128 %0, %1, off"
               :: "v"(lds_aw), "v"(g_aw) : "memory");
  asm volatile("global_load_async_to_lds_b128 %0, %1, off"
               :: "v"(lds_fk), "v"(g_fk) : "memory");
  asm volatile("global_load_async_to_lds_b128 %0, %1, off offset:16"
               :: "v"(lds_fk), "v"(g_fk) : "memory");
}

__global__ __launch_bounds__(32 * WAVES_PER_BLOCK)
void decomposed_embedding_gather_kernel(const int* __restrict__ ids,
                                        const float* __restrict__ sw,
                                        const float* __restrict__ mask,
                                        const float* __restrict__ aw,
                                        const float* __restrict__ atten,
                                        const float* __restrict__ fkb,
                                        float* __restrict__ out,
                                        int n_tokens) {
  __shared__ __align__(16) unsigned char smem[WAVES_PER_BLOCK * 2 * STAGE_BYTES];

  const int lane = threadIdx.x & 31;
  const int wid  = threadIdx.x >> 5;
  const int wave = blockIdx.x * (blockDim.x >> 5) + wid;
  const int wstride = gridDim.x * (blockDim.x >> 5);

  const float a0 = atten[0];
  const float a1 = atten[1];

  unsigned char* wbuf = &smem[wid * 2 * STAGE_BYTES];
  // Low 32 bits of the flat pointer == workgroup-relative LDS byte offset,
  // which is exactly what the async engine's VDST operand expects.
  const uint32_t wbase = (uint32_t)(uintptr_t)wbuf;

  uint32_t lds_sw[2], lds_aw[2], lds_fk[2];
#pragma unroll
  for (int st = 0; st < 2; ++st) {
    const uint32_t b = wbase + (uint32_t)st * STAGE_BYTES;
    lds_sw[st] = b + (uint32_t)lane * 16u;
    lds_aw[st] = b + 512u + (uint32_t)lane * 16u;
    lds_fk[st] = b + 1024u + (uint32_t)lane * 32u;
  }

  int cur = wave;
  if (cur >= n_tokens) return;

  const uint64_t sw_b  = (uint64_t)(uintptr_t)sw;
  const uint64_t aw_b  = (uint64_t)(uintptr_t)aw;
  const uint64_t fkb_b = (uint64_t)(uintptr_t)fkb;
  const uint64_t lsw = (uint64_t)lane * 16u;
  const uint64_t lfk = (uint64_t)lane * 32u;

  float mcur, mnxt = 0.0f;
  {
    const int idx = ids[cur];
    mcur = mask[idx];
    const uint64_t row = (uint64_t)(uint32_t)idx;
    async_issue_row(lds_sw[0], lds_aw[0], lds_fk[0],
                    sw_b  + row * ROW_BYTES_SWAW + lsw,
                    aw_b  + row * ROW_BYTES_SWAW + lsw,
                    fkb_b + row * ROW_BYTES_FKB  + lfk);
  }

  int s = 0;
  for (;;) {
    const int nxt = cur + wstride;
    const bool have_next = (nxt < n_tokens);

    if (have_next) {
      const int idx = ids[nxt];
      mnxt = mask[idx];
      const uint64_t row = (uint64_t)(uint32_t)idx;
      // DS reads and async-engine LDS writes complete on unordered paths:
      // make sure the previous consumption of buffer s^1 has fully retired
      // before the async engine is allowed to overwrite it.
      asm volatile("s_wait_dscnt 0x0" ::: "memory");
      async_issue_row(lds_sw[s ^ 1], lds_aw[s ^ 1], lds_fk[s ^ 1],
                      sw_b  + row * ROW_BYTES_SWAW + lsw,
                      aw_b  + row * ROW_BYTES_SWAW + lsw,
                      fkb_b + row * ROW_BYTES_FKB  + lfk);
      // 8 async ops outstanding; loads complete in order, so <=4 means the
      // current stage's 4 transfers have landed in LDS.
      asm volatile("s_wait_asynccnt 0x4" ::: "memory");
    } else {
      asm volatile("s_wait_asynccnt 0x0" ::: "memory");
    }

    const unsigned char* sb = wbuf + s * STAGE_BYTES;
    const float4 sv = *(const float4*)(sb + lane * 16);
    const float4 av = *(const float4*)(sb + 512 + lane * 16);
    const float4 f0 = *(const float4*)(sb + 1024 + lane * 32);
    const float4 f1 = *(const float4*)(sb + 1024 + lane * 32 + 16);

    const float sig = 1.0f / (1.0f + __expf(-mcur));

    float4 o;
    o.x = fmaf(sv.x, sig, av.x) + fmaf(a0, f0.x, a1 * f0.y);
    o.y = fmaf(sv.y, sig, av.y) + fmaf(a0, f0.z, a1 * f0.w);
    o.z = fmaf(sv.z, sig, av.z) + fmaf(a0, f1.x, a1 * f1.y);
    o.w = fmaf(sv.w, sig, av.w) + fmaf(a0, f1.z, a1 * f1.w);

    *(float4*)(out + (size_t)cur * EMBED_DIM + (size_t)lane * 4) = o;

    if (!have_next) break;
    mcur = mnxt;
    cur = nxt;
    s ^= 1;
  }
}

extern "C" void kernel_launch(void* const* d_in, const int* in_sizes, int n_in,
                              void* d_out, int out_size, void* d_ws, size_t ws_size,
                              hipStream_t stream) {
  (void)n_in; (void)out_size; (void)d_ws; (void)ws_size;

  const int*   ids   = (const int*)d_in[0];    // input_ids [32*512]
  const float* sw    = (const float*)d_in[1];  // [V,128]
  const float* mask  = (const float*)d_in[2];  // [V]
  const float* aw    = (const float*)d_in[3];  // [V,128]
  const float* atten = (const float*)d_in[4];  // [2]
  const float* fkb   = (const float*)d_in[5];  // [V,128,2]
  float*       out   = (float*)d_out;          // [32*512*128]

  const int n_tokens = in_sizes[0];            // 16384

  // ~4 tokens per wave => pipeline depth matters, machine stays saturated.
  int waves = (n_tokens + 3) / 4;
  int grid = (waves + WAVES_PER_BLOCK - 1) / WAVES_PER_BLOCK;
  if (grid < 1) grid = 1;
  if (grid > 16384) grid = 16384;

  decomposed_embedding_gather_kernel<<<grid, dim3(32 * WAVES_PER_BLOCK), 0, stream>>>(
      ids, sw, mask, aw, atten, fkb, out, n_tokens);
}